// MaceNet_64879775973535
// MI455X (gfx1250) — compile-verified
//
#include <hip/hip_runtime.h>
#include <math.h>

// MACE-like equivariant GNN for MI455X (gfx1250, wave32, WMMA).
// All GEMM-shaped work (edge MLP, aggregation, node chain, outputs) runs on
// V_WMMA_F32_16X16X4_F32 (fp32-faithful path).  Edge-MLP weights are staged
// in LDS once per block; aggregation prefetches its strided streams.

#define NN 256              // atoms per batch
#define BB 4                // batches
#define FF 32               // features
#define EE (NN * (NN - 1))  // 65280 edges per batch (all pairs, no self)
#define RMAXF 10.0f

typedef __attribute__((ext_vector_type(2))) float v2f;
typedef __attribute__((ext_vector_type(8))) float v8f;

// Fast silu: x * v_rcp_f32(1 + exp(-x))  (avoids the IEEE div expansion)
__device__ __forceinline__ float silu_f(float x) {
  return x * __builtin_amdgcn_rcpf(1.0f + __expf(-x));
}

__device__ __forceinline__ v8f v8_splat(float x) {
  v8f r;
  for (int i = 0; i < 8; ++i) r[i] = x;
  return r;
}

// Generic 16x16 fp32 WMMA tile accumulate: acc += A(16xK) @ B(Kx16).
// Compile-time strides; contiguous-A path uses a single b64 load per step
// (alignment: even row strides, even ka, 8B-aligned bases at all call sites).
// Fragment layouts per CDNA5 ISA 7.12.2.
template <int LDA_R, int LDA_C, int LDB_R, int LDB_C>
__device__ __forceinline__ v8f wmma_k(const float* A, const float* Bm, int K,
                                      v8f acc) {
  const int lane = threadIdx.x & 31;
  const int half = lane >> 4;
  const int l16 = lane & 15;
  for (int k = 0; k < K; k += 4) {
    const int ka = k + 2 * half;
    v2f a, b;
    if constexpr (LDA_C == 1) {
      a = *reinterpret_cast<const v2f*>(A + l16 * LDA_R + ka);
    } else {
      a[0] = A[l16 * LDA_R + (ka + 0) * LDA_C];
      a[1] = A[l16 * LDA_R + (ka + 1) * LDA_C];
    }
    b[0] = Bm[(ka + 0) * LDB_R + l16 * LDB_C];
    b[1] = Bm[(ka + 1) * LDB_R + l16 * LDB_C];
    acc = __builtin_amdgcn_wmma_f32_16x16x4_f32(false, a, false, b, (short)0,
                                                acc, false, false);
  }
  return acc;
}

// ---------------------------------------------------------------- init ------
__global__ void k_init(const int* __restrict__ h, const float* __restrict__ emb,
                       float* __restrict__ hf, float* __restrict__ vf) {
  int i = blockIdx.x * blockDim.x + threadIdx.x;
  if (i < BB * NN * FF) {
    int f = i % FF;
    int n = (i / FF) % NN;
    int b = i / (NN * FF);
    hf[i] = emb[h[b * NN + n] * FF + f];
  }
  if (i < BB * 3 * NN * FF) vf[i] = 0.0f;
}

// ------------------------------------------------- edge geometry (Y, bessel)
__global__ void k_geom(const float* __restrict__ x, float* __restrict__ Y,
                       float* __restrict__ rb) {
  int g = blockIdx.x * blockDim.x + threadIdx.x;
  if (g >= BB * EE) return;
  int b = g / EE, e = g % EE;
  int s = e / (NN - 1), j = e % (NN - 1);
  int r = j + (j >= s ? 1 : 0);
  const float* ps = x + (size_t)(b * NN + s) * 3;
  const float* pr = x + (size_t)(b * NN + r) * 3;
  float vx = pr[0] - ps[0], vy = pr[1] - ps[1], vz = pr[2] - ps[2];
  float d = sqrtf(vx * vx + vy * vy + vz * vz);
  float inv = 1.0f / d;
  float ux = vx * inv, uy = vy * inv, uz = vz * inv;
  const float s3 = 1.7320508075688772f;   // sqrt(3)
  const float s15 = 3.872983346207417f;   // sqrt(15)
  const float s5h = 1.118033988749895f;   // sqrt(5)/2
  float* y = Y + (size_t)g * 9;
  y[0] = 1.0f;
  y[1] = s3 * ux;
  y[2] = s3 * uy;
  y[3] = s3 * uz;
  y[4] = s15 * ux * uy;
  y[5] = s15 * uy * uz;
  y[6] = s5h * (3.0f * uz * uz - 1.0f);
  y[7] = s15 * ux * uz;
  y[8] = 0.5f * s15 * (ux * ux - uy * uy);
  float* rbp = rb + (size_t)g * 8;
  const float kb = 0.4472135954999579f;  // sqrt(2/RMAX)
  const float w = 3.14159265358979323846f / RMAXF;
#pragma unroll
  for (int n = 1; n <= 8; ++n) rbp[n - 1] = kb * __sinf((float)n * w * d) * inv;
}

// ------------------------------------- edge MLP: R' = hf[s] * (silu MLP(rb))
// 8 waves / block, one 16-edge tile per wave.  rW1/rW2/biases staged in LDS
// once per block (kills the redundant per-tile global weight re-fetch);
// GEMM1 8->64 (silu, LDS re-layout D->A), GEMM2 64->32, scale by hf[sender].
__global__ void __launch_bounds__(256) k_edge_mlp(
    const float* __restrict__ rb, const float* __restrict__ rW1,
    const float* __restrict__ rb1, const float* __restrict__ rW2,
    const float* __restrict__ rb2, const float* __restrict__ hf,
    float* __restrict__ Rp) {
  __shared__ float sW1[8 * 64];    // 512
  __shared__ float sW2[64 * 32];   // 2048
  __shared__ float sB1[64];
  __shared__ float sB2[32];
  __shared__ float hid[8][16 * 64];

  const int tid = threadIdx.x;
  // cooperative weight staging
  for (int i = tid; i < 8 * 64; i += 256) sW1[i] = rW1[i];
  for (int i = tid; i < 64 * 32; i += 256) sW2[i] = rW2[i];
  if (tid < 64) sB1[tid] = rb1[tid];
  if (tid < 32) sB2[tid] = rb2[tid];
  __syncthreads();

  const int lane = tid & 31;
  const int w = tid >> 5;
  const int tile = blockIdx.x * 8 + w;  // 0 .. BB*EE/16-1
  const int b = tile / (EE / 16);
  const int t = tile % (EE / 16);
  const int ebase = t * 16;
  const size_t g = (size_t)b * EE + ebase;
  const int half = lane >> 4, l16 = lane & 15;

  const float* Atile = rb + g * 8;  // (16 x 8) edge-tile of bessel features
  for (int nt = 0; nt < 4; ++nt) {
    v8f acc = v8_splat(sB1[nt * 16 + l16]);
    acc = wmma_k<8, 1, 64, 1>(Atile, sW1 + nt * 16, 8, acc);
#pragma unroll
    for (int v = 0; v < 8; ++v) {
      int M = v + 8 * half;
      hid[w][M * 64 + nt * 16 + l16] = silu_f(acc[v]);
    }
  }
  asm volatile("s_wait_dscnt 0" ::: "memory");  // LDS W->R within wave

  for (int nt = 0; nt < 2; ++nt) {
    v8f acc = v8_splat(sB2[nt * 16 + l16]);
    acc = wmma_k<64, 1, 32, 1>(hid[w], sW2 + nt * 16, 64, acc);
#pragma unroll
    for (int v = 0; v < 8; ++v) {
      int M = v + 8 * half;
      int snd = (ebase + M) / (NN - 1);  // sender of this edge
      float hv = hf[(size_t)(b * NN + snd) * FF + nt * 16 + l16];
      Rp[(g + M) * FF + nt * 16 + l16] = acc[v] * hv;
    }
  }
}

// ----------------- aggregation as WMMA: agg[n] = (1/N) P_n^T @ Y_n ----------
// Dense all-pairs graph: per receiver n the segment-sum is a (32 x 256) @
// (256 x 16-padded) GEMM with K = N exactly.  One wave per receiver, two
// 16x16 accumulators (F = 32 rows), diagonal s==n zero-masked on the B
// fragment (edge index clamped to a valid address; EXEC stays full).
// global_prefetch one K-step ahead on the strided R'/Y streams.
__global__ void __launch_bounds__(128) k_agg(const float* __restrict__ Rp,
                                             const float* __restrict__ Y,
                                             float* __restrict__ agg) {
  const int lane = threadIdx.x & 31;
  const int w = threadIdx.x >> 5;
  const int rcv = blockIdx.x * 4 + w;  // 0 .. BB*NN-1
  const int b = rcv / NN, n = rcv % NN;
  const int half = lane >> 4, l16 = lane & 15;
  const size_t gb = (size_t)b * EE;

  v8f acc0 = v8_splat(0.0f), acc1 = v8_splat(0.0f);
  for (int k = 0; k < NN; k += 4) {
    const int s0 = k + 2 * half, s1 = s0 + 1;
    const int e0 = (s0 == n) ? 0 : (s0 * (NN - 1) + (n > s0 ? n - 1 : n));
    const int e1 = (s1 == n) ? 0 : (s1 * (NN - 1) + (n > s1 ? n - 1 : n));
    const float m0 = (s0 == n) ? 0.0f : 1.0f;
    const float m1 = (s1 == n) ? 0.0f : 1.0f;
    const size_t g0 = gb + e0, g1 = gb + e1;
    // prefetch next K-step (analytic edge index, +4 senders ahead)
    if (k + 4 < NN) {
      const int sp = s0 + 4;
      const int ep = (sp == n) ? 0 : (sp * (NN - 1) + (n > sp ? n - 1 : n));
      const size_t gp = gb + ep;
      __builtin_prefetch(Rp + gp * FF + l16, 0, 1);
      __builtin_prefetch(Y + gp * 9 + (l16 < 9 ? l16 : 0), 0, 1);
    }
    v2f bv, a0, a1;
    bv[0] = m0 * ((l16 < 9) ? Y[g0 * 9 + l16] : 0.0f);
    bv[1] = m1 * ((l16 < 9) ? Y[g1 * 9 + l16] : 0.0f);
    a0[0] = Rp[g0 * FF + l16];
    a0[1] = Rp[g1 * FF + l16];
    a1[0] = Rp[g0 * FF + 16 + l16];
    a1[1] = Rp[g1 * FF + 16 + l16];
    acc0 = __builtin_amdgcn_wmma_f32_16x16x4_f32(false, a0, false, bv, (short)0,
                                                 acc0, false, false);
    acc1 = __builtin_amdgcn_wmma_f32_16x16x4_f32(false, a1, false, bv, (short)0,
                                                 acc1, false, false);
  }
  if (l16 < 9) {
    float* ag = agg + (size_t)rcv * 288;
#pragma unroll
    for (int v = 0; v < 8; ++v) {
      int f = v + 8 * half;
      ag[f * 9 + l16] = acc0[v] * (1.0f / NN);
      ag[(16 + f) * 9 + l16] = acc1[v] * (1.0f / NN);
    }
  }
}

// ------------------- node update chain: Wcomb, poly(A), res-MLP, hf+=, vf+= -
// One wave per 16-node tile; whole chain stays in registers/LDS.
__global__ void __launch_bounds__(32) k_node(
    const float* __restrict__ agg, float* __restrict__ hf,
    float* __restrict__ vf, const float* __restrict__ Wcomb,
    const float* __restrict__ Amat, const float* __restrict__ rsW1,
    const float* __restrict__ rsb1, const float* __restrict__ rsW2,
    const float* __restrict__ rsb2, const float* __restrict__ Wv) {
  __shared__ float sA[16 * 64];
  __shared__ float sF[16 * 32], sF2[16 * 32], sF3[16 * 32], sC[16 * 32];
  __shared__ float sH[16 * 64];
  const int lane = threadIdx.x & 31;
  const int b = blockIdx.x >> 4, tile = blockIdx.x & 15;
  const int nbase = tile * 16;
  const int half = lane >> 4, l16 = lane & 15;

  // sA = [ s0 | i2 ]  (16 x 64):  s0 = agg[...,0],  i2 = sum_c agg[...,4+c]^2
  for (int idx = lane; idx < 16 * 64; idx += 32) {
    int row = idx >> 6, col = idx & 63;
    const float* ag = agg + (size_t)(b * NN + nbase + row) * 288;
    float val;
    if (col < 32) {
      val = ag[col * 9];
    } else {
      int fe = col - 32;
      val = 0.0f;
#pragma unroll
      for (int c2 = 0; c2 < 5; ++c2) {
        float tv = ag[fe * 9 + 4 + c2];
        val += tv * tv;
      }
    }
    sA[idx] = val;
  }
  asm volatile("s_wait_dscnt 0" ::: "memory");

  // f = sA @ Wcomb   (16x64)@(64x32);  also stage f^2, f^3
  for (int nt = 0; nt < 2; ++nt) {
    v8f acc = v8_splat(0.0f);
    acc = wmma_k<64, 1, 32, 1>(sA, Wcomb + nt * 16, 64, acc);
#pragma unroll
    for (int v = 0; v < 8; ++v) {
      int o = (v + 8 * half) * 32 + nt * 16 + l16;
      float fv = acc[v];
      sF[o] = fv;
      sF2[o] = fv * fv;
      sF3[o] = fv * fv * fv;
    }
  }
  asm volatile("s_wait_dscnt 0" ::: "memory");

  // c = f@A0 + f^2@A1 + f^3@A2  (chained WMMA accumulation)
  for (int nt = 0; nt < 2; ++nt) {
    v8f acc = v8_splat(0.0f);
    acc = wmma_k<32, 1, 32, 1>(sF, Amat + 0 * 32 * 32 + nt * 16, 32, acc);
    acc = wmma_k<32, 1, 32, 1>(sF2, Amat + 1 * 32 * 32 + nt * 16, 32, acc);
    acc = wmma_k<32, 1, 32, 1>(sF3, Amat + 2 * 32 * 32 + nt * 16, 32, acc);
#pragma unroll
    for (int v = 0; v < 8; ++v)
      sC[(v + 8 * half) * 32 + nt * 16 + l16] = acc[v];
  }
  asm volatile("s_wait_dscnt 0" ::: "memory");

  // h1 = silu(c @ resW1 + resb1)   (16x32)@(32x64)
  for (int nt = 0; nt < 4; ++nt) {
    v8f acc = v8_splat(rsb1[nt * 16 + l16]);
    acc = wmma_k<32, 1, 64, 1>(sC, rsW1 + nt * 16, 32, acc);
#pragma unroll
    for (int v = 0; v < 8; ++v)
      sH[(v + 8 * half) * 64 + nt * 16 + l16] = silu_f(acc[v]);
  }
  asm volatile("s_wait_dscnt 0" ::: "memory");

  // hf += h1 @ resW2 + resb2   (16x64)@(64x32)
  for (int nt = 0; nt < 2; ++nt) {
    v8f acc = v8_splat(rsb2[nt * 16 + l16]);
    acc = wmma_k<64, 1, 32, 1>(sH, rsW2 + nt * 16, 64, acc);
#pragma unroll
    for (int v = 0; v < 8; ++v) {
      int M = v + 8 * half;
      size_t o = (size_t)(b * NN + nbase + M) * FF + nt * 16 + l16;
      hf[o] += acc[v];
    }
  }

  // vf[:,c] += v1[:,:,c] @ Wv   — A loaded strided straight from agg
  for (int c3 = 0; c3 < 3; ++c3) {
    const float* Av = agg + (size_t)(b * NN + nbase) * 288 + 1 + c3;
    for (int nt = 0; nt < 2; ++nt) {
      v8f acc = v8_splat(0.0f);
      acc = wmma_k<288, 9, 32, 1>(Av, Wv + nt * 16, 32, acc);
#pragma unroll
      for (int v = 0; v < 8; ++v) {
        int M = v + 8 * half;
        size_t o =
            (size_t)((b * 3 + c3) * NN + nbase + M) * FF + nt * 16 + l16;
        vf[o] += acc[v];
      }
    }
  }
}

// ------------------------------------------------ output projections --------
__global__ void __launch_bounds__(32) k_out(
    const float* __restrict__ hf, const float* __restrict__ vf,
    const float* __restrict__ Wvec, const float* __restrict__ Winv,
    const float* __restrict__ binv, float* __restrict__ out) {
  const int lane = threadIdx.x & 31;
  const int b = blockIdx.x >> 4, tile = blockIdx.x & 15;
  const int nbase = tile * 16;
  const int half = lane >> 4, l16 = lane & 15;

  // iout = hf @ Winv_out + binv   (16x32)@(32x64)
  const float* Ah = hf + (size_t)(b * NN + nbase) * FF;
  float* iout = out + (size_t)BB * NN * 96 + (size_t)b * NN * 64;
  for (int nt = 0; nt < 4; ++nt) {
    v8f acc = v8_splat(binv[nt * 16 + l16]);
    acc = wmma_k<32, 1, 64, 1>(Ah, Winv + nt * 16, 32, acc);
#pragma unroll
    for (int v = 0; v < 8; ++v) {
      int M = v + 8 * half;
      iout[(size_t)(nbase + M) * 64 + nt * 16 + l16] = acc[v];
    }
  }

  // vout[:, :, c] = vf[:, :, c] @ Wvec_out   (16x32)@(32x32), col stride 3
  for (int c3 = 0; c3 < 3; ++c3) {
    const float* Av = vf + (size_t)((b * 3 + c3) * NN + nbase) * FF;
    for (int nt = 0; nt < 2; ++nt) {
      v8f acc = v8_splat(0.0f);
      acc = wmma_k<32, 1, 32, 1>(Av, Wvec + nt * 16, 32, acc);
#pragma unroll
      for (int v = 0; v < 8; ++v) {
        int M = v + 8 * half;
        out[(size_t)b * NN * 96 + (size_t)(nbase + M) * 96 +
            (nt * 16 + l16) * 3 + c3] = acc[v];
      }
    }
  }
}

// ---------------------------------------------------------------------------
extern "C" void kernel_launch(void* const* d_in, const int* in_sizes, int n_in,
                              void* d_out, int out_size, void* d_ws,
                              size_t ws_size, hipStream_t stream) {
  const float* x = (const float*)d_in[0];
  const int* h = (const int*)d_in[1];
  const float* emb = (const float*)d_in[2];
  const float* rW1 = (const float*)d_in[3];
  const float* rb1 = (const float*)d_in[4];
  const float* rW2 = (const float*)d_in[5];
  const float* rb2 = (const float*)d_in[6];
  const float* Wcomb = (const float*)d_in[7];
  const float* Amat = (const float*)d_in[8];
  const float* resW1 = (const float*)d_in[9];
  const float* resb1 = (const float*)d_in[10];
  const float* resW2 = (const float*)d_in[11];
  const float* resb2 = (const float*)d_in[12];
  const float* Wv = (const float*)d_in[13];
  const float* Wvec = (const float*)d_in[14];
  const float* Winv = (const float*)d_in[15];
  const float* binv = (const float*)d_in[16];
  float* out = (float*)d_out;

  // workspace carve-up (floats); total ~13.2M floats (~53 MB)
  float* ws = (float*)d_ws;
  float* Y = ws;
  ws += (size_t)BB * EE * 9;
  float* rbw = ws;
  ws += (size_t)BB * EE * 8;
  float* Rp = ws;
  ws += (size_t)BB * EE * FF;
  float* agg = ws;
  ws += (size_t)BB * NN * 288;
  float* hf = ws;
  ws += (size_t)BB * NN * FF;
  float* vf = ws;
  ws += (size_t)BB * 3 * NN * FF;

  k_init<<<(BB * 3 * NN * FF + 255) / 256, 256, 0, stream>>>(h, emb, hf, vf);
  k_geom<<<(BB * EE + 255) / 256, 256, 0, stream>>>(x, Y, rbw);
  for (int l = 0; l < 2; ++l) {
    k_edge_mlp<<<(BB * (EE / 16)) / 8, 256, 0, stream>>>(
        rbw, rW1 + l * 8 * 64, rb1 + l * 64, rW2 + l * 64 * 32, rb2 + l * 32,
        hf, Rp);
    k_agg<<<(BB * NN) / 4, 128, 0, stream>>>(Rp, Y, agg);
    k_node<<<BB * 16, 32, 0, stream>>>(
        agg, hf, vf, Wcomb + l * 64 * 32, Amat + l * 3 * 32 * 32,
        resW1 + l * 32 * 64, resb1 + l * 64, resW2 + l * 64 * 32,
        resb2 + l * 32, Wv + l * 32 * 32);
  }
  k_out<<<BB * 16, 32, 0, stream>>>(hf, vf, Wvec, Winv, binv, out);
}